// FACL_4123168604382
// MI455X (gfx1250) — compile-verified
//
#include <hip/hip_runtime.h>
#include <hip/hip_bf16.h>
#include <math.h>

// ---------------------------------------------------------------------------
// Problem constants (from reference)
// ---------------------------------------------------------------------------
#define NN      1024
#define X1_DIM  512
#define X2_DIM  768
#define Y_DIM   10
#define HID     128
#define EMB     64
#define ALPHA   0.0001f
#define INV_TEMP 10.0f   // 1/0.1

typedef __attribute__((ext_vector_type(2))) float v2f;
typedef __attribute__((ext_vector_type(8))) float v8f;

// ---------------------------------------------------------------------------
// s_wait_asynccnt helper (builtin if present, else raw encoding via asm)
// ---------------------------------------------------------------------------
__device__ __forceinline__ void facl_wait_async0()
{
#if defined(__has_builtin) && __has_builtin(__builtin_amdgcn_s_wait_asynccnt)
    __builtin_amdgcn_s_wait_asynccnt(0);
#else
    asm volatile("s_wait_asynccnt 0x0" ::: "memory");
#endif
}

// ---------------------------------------------------------------------------
// Fused GEMM + bias + (optional ReLU) using V_WMMA_F32_16X16X4_F32.
// Out[M,N] = act(X[M,K] @ W[K,N] + bias[N]); all row-major fp32.
// One wave per 16x16 output tile, 4 waves (4 M-tiles) per block.
// grid = (M/64, N/16), block = 128.
// Strength-reduced addressing: A fragment via one 8B load, B via two
// coalesced b32 loads; pointers bumped by constants per K-step.
// ---------------------------------------------------------------------------
__global__ __launch_bounds__(128) void facl_gemm_wmma(
    const float* __restrict__ X, const float* __restrict__ W,
    const float* __restrict__ bias, float* __restrict__ Out,
    int M, int K, int N, int act)
{
    const int lane = threadIdx.x & 31;
    const int wave = threadIdx.x >> 5;
    const int half = lane >> 4;      // 0/1 -> K sub-pair
    const int lr   = lane & 15;
    const int m0   = (blockIdx.x * 4 + wave) * 16;
    const int n0   = blockIdx.y * 16;

    v8f acc = {};
    const float* ap = X + (size_t)(m0 + lr) * K + 2 * half;   // 8B aligned
    const float* bp = W + (size_t)(2 * half) * N + n0 + lr;

    for (int k = 0; k < K; k += 4) {
        const v2f a = *(const v2f*)ap;          // A[m, ka], A[m, ka+1]
        v2f b;
        b.x = bp[0];                             // B[ka,   n]
        b.y = bp[N];                             // B[ka+1, n]
        acc = __builtin_amdgcn_wmma_f32_16x16x4_f32(
            false, a, false, b, (short)0, acc, false, false);
        ap += 4;
        bp += (size_t)4 * N;
    }

    const float bv = bias ? bias[n0 + lr] : 0.0f;
#pragma unroll
    for (int r = 0; r < 8; ++r) {
        float v = acc[r] + bv;
        if (act) v = fmaxf(v, 0.0f);
        Out[(size_t)(m0 + r + 8 * half) * N + n0 + lr] = v;
    }
}

// ---------------------------------------------------------------------------
// Row L2-normalize for [rows, 64] in place. grid = rows, block = 64.
// ---------------------------------------------------------------------------
__global__ __launch_bounds__(64) void facl_rownorm64(float* __restrict__ e)
{
    __shared__ float red[64];
    const int row = blockIdx.x, t = threadIdx.x;
    const float v = e[row * EMB + t];
    red[t] = v * v;
    __syncthreads();
    for (int s = 32; s > 0; s >>= 1) {
        if (t < s) red[t] += red[t + s];
        __syncthreads();
    }
    const float nrm = fmaxf(sqrtf(red[0]), 1e-12f);
    e[row * EMB + t] = v / nrm;
}

// ---------------------------------------------------------------------------
// CLUB one-hot fixup: hx_c[i,h] += club_w1[(EMB + y[i]) * HID + h]
// ---------------------------------------------------------------------------
__global__ __launch_bounds__(HID) void facl_onehot_add(
    float* __restrict__ hx, const int* __restrict__ y,
    const float* __restrict__ club_w1)
{
    const int i = blockIdx.x, h = threadIdx.x;
    hx[i * HID + h] += club_w1[(size_t)(EMB + y[i]) * HID + h];
}

// ---------------------------------------------------------------------------
// Pair-score reduction:
//   s[i,j] = (sum_h relu(hx[j,h] + hy[i,h] + b1[h]) * w2[h] + b2) / TEMP
// Streams over j with DOUBLE-BUFFERED async DMA of hx tiles into LDS
// (GLOBAL_LOAD_ASYNC_TO_LDS_B128, ASYNCcnt), never materializing s.
// Per-block (8 rows) outputs:
//   out_lse[b]  = sum_i logsumexp_j s[i,j]
//   out_diag[b] = sum_i s[i,i]
//   out_all[b]  = sum_ij s[i,j]
// grid = NN/8, block = 256 (8 waves, one wave per row i).
// HPAD=132: rows 528B -> 16B aligned; wave-wide ds_load_b128 hits each
// of the 64 LDS banks exactly twice (minimum possible).
// ---------------------------------------------------------------------------
#define TI 8
#define TJ 32
#define HPAD 132
#define NTJ (NN / TJ)

__global__ __launch_bounds__(256) void facl_pair_reduce(
    const float* __restrict__ hx, const float* __restrict__ hy,
    const float* __restrict__ b1, const float* __restrict__ w2,
    const float* __restrict__ b2p,
    float* __restrict__ out_lse, float* __restrict__ out_diag,
    float* __restrict__ out_all)
{
    __shared__ __align__(16) float sh_hx[2][TJ][HPAD];
    __shared__ __align__(16) float sh_hy[TI][HPAD];
    __shared__ __align__(16) float sh_w2[HID];
    __shared__ float red_m[TI], red_t[TI], red_d[TI], red_a[TI];

    const int tid = threadIdx.x;
    const int ti = tid >> 5;        // wave id == local row
    const int tj = tid & 31;        // lane id == local col
    const int i0 = blockIdx.x * TI;
    const int i  = i0 + ti;

    // Issue one 16KB hx j-tile (32 rows x 128 floats) as 1024 async b128
    // DMAs into LDS: 4 chunks per thread.
    auto issue_tile = [&](int j0, int buf) {
#pragma unroll
        for (int it = 0; it < 4; ++it) {
            const int c = tid + 256 * it;     // chunk id 0..1023
            const int r = c >> 5;             // 32 chunks per row
            const int h = (c & 31) << 2;      // float offset in row
            const unsigned lds = (unsigned)(uintptr_t)&sh_hx[buf][r][h];
            const int voff = ((j0 + r) * HID + h) * 4;  // byte offset
            asm volatile("global_load_async_to_lds_b128 %0, %1, %2"
                         :: "v"(lds), "v"(voff), "s"(hx)
                         : "memory");
        }
    };

    // Stage hy (+b1 folded) and w2 with normal loads.
    for (int idx = tid; idx < TI * HID; idx += 256) {
        const int r = idx >> 7, h = idx & (HID - 1);
        sh_hy[r][h] = hy[(i0 + r) * HID + h] + b1[h];
    }
    if (tid < HID) sh_w2[tid] = w2[tid];
    const float b2 = b2p[0];

    issue_tile(0, 0);
    facl_wait_async0();
    __syncthreads();

    float m = -INFINITY, t = 0.0f, dsum = 0.0f, asum = 0.0f;
    int cur = 0;

    for (int jt = 0; jt < NTJ; ++jt) {
        const int nxt = cur ^ 1;
        if (jt + 1 < NTJ) issue_tile((jt + 1) * TJ, nxt);  // overlap DMA

        float acc = 0.0f;
#pragma unroll
        for (int h = 0; h < HID; h += 4) {
            const float4 xv = *(const float4*)&sh_hx[cur][tj][h];
            const float4 yv = *(const float4*)&sh_hy[ti][h];
            const float4 wv = *(const float4*)&sh_w2[h];
            acc += fmaxf(xv.x + yv.x, 0.0f) * wv.x;
            acc += fmaxf(xv.y + yv.y, 0.0f) * wv.y;
            acc += fmaxf(xv.z + yv.z, 0.0f) * wv.z;
            acc += fmaxf(xv.w + yv.w, 0.0f) * wv.w;
        }
        const float s = (acc + b2) * INV_TEMP;
        if (jt * TJ + tj == i) dsum += s;
        asum += s;
        if (s > m) { t = t * __expf(m - s) + 1.0f; m = s; }
        else       { t += __expf(s - m); }

        facl_wait_async0();   // own wave's DMAs into nxt done
        __syncthreads();      // all waves done issuing nxt + reading cur
        cur = nxt;
    }

    // Merge the 32 streaming-LSE states of one row (one wave).
    for (int off = 16; off >= 1; off >>= 1) {
        const float m2 = __shfl_xor(m, off, 32);
        const float t2 = __shfl_xor(t, off, 32);
        const float M  = fmaxf(m, m2);
        t = t * __expf(m - M) + t2 * __expf(m2 - M);
        m = M;
        dsum += __shfl_xor(dsum, off, 32);
        asum += __shfl_xor(asum, off, 32);
    }
    if (tj == 0) { red_m[ti] = m; red_t[ti] = t; red_d[ti] = dsum; red_a[ti] = asum; }
    __syncthreads();
    if (tid == 0) {
        float lse = 0.0f, dd = 0.0f, aa = 0.0f;
        for (int r = 0; r < TI; ++r) {
            lse += red_m[r] + __logf(red_t[r]);
            dd  += red_d[r];
            aa  += red_a[r];
        }
        out_lse[blockIdx.x]  = lse;
        out_diag[blockIdx.x] = dd;
        out_all[blockIdx.x]  = aa;
    }
}

// ---------------------------------------------------------------------------
// Deterministic final reduction of per-block partials -> scalar loss.
// part layout: [0:128) nce lse | [128:256) nce diag | [256:384) nce all
//              [384:512) club lse | [512:640) club diag | [640:768) club all
// ---------------------------------------------------------------------------
__global__ void facl_finalize(const float* __restrict__ part, float* __restrict__ out)
{
    if (threadIdx.x != 0 || blockIdx.x != 0) return;
    float lse = 0.0f, pos = 0.0f, dc = 0.0f, ac = 0.0f;
    for (int b = 0; b < NN / TI; ++b) {
        lse += part[b];
        pos += part[128 + b];
        dc  += part[512 + b];
        ac  += part[640 + b];
    }
    const float n = (float)NN;
    const float loss_uncond = lse / n - logf(n) - pos / n;
    const float loss_cond   = dc / n - ac / (n * n);
    out[0] = ALPHA * (loss_uncond + loss_cond);
}

// ---------------------------------------------------------------------------
// Host-side launch
// ---------------------------------------------------------------------------
extern "C" void kernel_launch(void* const* d_in, const int* in_sizes, int n_in,
                              void* d_out, int out_size, void* d_ws, size_t ws_size,
                              hipStream_t stream)
{
    const float* x1      = (const float*)d_in[0];
    const float* x2      = (const float*)d_in[1];
    const int*   y       = (const int*)  d_in[2];
    const float* enc1_w1 = (const float*)d_in[3];
    const float* enc1_b1 = (const float*)d_in[4];
    const float* enc1_w2 = (const float*)d_in[5];
    const float* enc1_b2 = (const float*)d_in[6];
    const float* enc2_w1 = (const float*)d_in[7];
    const float* enc2_b1 = (const float*)d_in[8];
    const float* enc2_w2 = (const float*)d_in[9];
    const float* enc2_b2 = (const float*)d_in[10];
    const float* nce_w1  = (const float*)d_in[11];
    const float* nce_b1  = (const float*)d_in[12];
    const float* nce_w2  = (const float*)d_in[13];
    const float* nce_b2  = (const float*)d_in[14];
    const float* club_w1 = (const float*)d_in[15];
    const float* club_b1 = (const float*)d_in[16];
    const float* club_w2 = (const float*)d_in[17];
    const float* club_b2 = (const float*)d_in[18];

    float* ws  = (float*)d_ws;
    float* x1e = ws;                       // 1024*64
    float* x2e = x1e + NN * EMB;           // 1024*64
    float* h1  = x2e + NN * EMB;           // 1024*128
    float* h2  = h1  + NN * HID;           // 1024*128
    float* hxn = h2  + NN * HID;           // 1024*128
    float* hyn = hxn + NN * HID;
    float* hxc = hyn + NN * HID;
    float* hyc = hxc + NN * HID;
    float* part = hyc + NN * HID;          // 768 floats of partials

    const dim3 blk128(128), blk64(64), blk256(256);

    // --- encoder 1: x1e = normalize(relu(x1@W1+b1)@W2+b2) ---
    facl_gemm_wmma<<<dim3(NN / 64, HID / 16), blk128, 0, stream>>>(
        x1, enc1_w1, enc1_b1, h1, NN, X1_DIM, HID, 1);
    facl_gemm_wmma<<<dim3(NN / 64, EMB / 16), blk128, 0, stream>>>(
        h1, enc1_w2, enc1_b2, x1e, NN, HID, EMB, 0);
    facl_rownorm64<<<NN, blk64, 0, stream>>>(x1e);

    // --- encoder 2 ---
    facl_gemm_wmma<<<dim3(NN / 64, HID / 16), blk128, 0, stream>>>(
        x2, enc2_w1, enc2_b1, h2, NN, X2_DIM, HID, 1);
    facl_gemm_wmma<<<dim3(NN / 64, EMB / 16), blk128, 0, stream>>>(
        h2, enc2_w2, enc2_b2, x2e, NN, HID, EMB, 0);
    facl_rownorm64<<<NN, blk64, 0, stream>>>(x2e);

    // --- NCE critic projections: hx = x1e@w1[:64], hy = x2e@w1[64:] ---
    facl_gemm_wmma<<<dim3(NN / 64, HID / 16), blk128, 0, stream>>>(
        x1e, nce_w1,              nullptr, hxn, NN, EMB, HID, 0);
    facl_gemm_wmma<<<dim3(NN / 64, HID / 16), blk128, 0, stream>>>(
        x2e, nce_w1 + EMB * HID,  nullptr, hyn, NN, EMB, HID, 0);

    // --- CLUB critic projections (one-hot part folded via row gather) ---
    facl_gemm_wmma<<<dim3(NN / 64, HID / 16), blk128, 0, stream>>>(
        x1e, club_w1,                         nullptr, hxc, NN, EMB, HID, 0);
    facl_onehot_add<<<NN, dim3(HID), 0, stream>>>(hxc, y, club_w1);
    facl_gemm_wmma<<<dim3(NN / 64, HID / 16), blk128, 0, stream>>>(
        x2e, club_w1 + (EMB + Y_DIM) * HID,   nullptr, hyc, NN, EMB, HID, 0);

    // --- streaming pair reductions (async-LDS double buffered) ---
    facl_pair_reduce<<<NN / TI, blk256, 0, stream>>>(
        hxn, hyn, nce_b1, nce_w2, nce_b2,
        part + 0, part + 128, part + 256);
    facl_pair_reduce<<<NN / TI, blk256, 0, stream>>>(
        hxc, hyc, club_b1, club_w2, club_b2,
        part + 384, part + 512, part + 640);

    // --- deterministic scalar combine ---
    facl_finalize<<<1, 1, 0, stream>>>(part, (float*)d_out);
}